// RandomParametricEQ_19739669692650
// MI455X (gfx1250) — compile-verified
//
#include <hip/hip_runtime.h>
#include <math.h>

typedef float v2f __attribute__((ext_vector_type(2)));
typedef float v8f __attribute__((ext_vector_type(8)));

#define NSTATE 10
#define BLK 16                 // samples per block
#define TILE 16                // blocks per tile (one wave32)
#define THREADS 1024           // 32 waves
#define BLOCKS_PER_CHUNK 512   // 32 waves * 16 blocks
#define CHUNK_SAMPS (BLOCKS_PER_CHUNK * BLK)   // 8192
#define CHUNKS_PER_SEG 4
#define SEG_SAMPS (CHUNKS_PER_SEG * CHUNK_SAMPS)  // 32768
#define NPOW 9                 // W^1..W^256 used by the in-chunk scan
#define NPOW_TOT 12            // chain extended to W^2048 for segment linking
#define ST_STRIDE 20           // floats per LDS state row (8B aligned, K<16 in-row)
#define NCH 8

// workspace layout (float offsets)
#define WS_T    0                      // 16x16 Toeplitz of cascade impulse response
#define WS_V    256                    // 16x16 (rows 0..9 = V, rest zero)
#define WS_U    512                    // 16x16 (cols 0..9 = U, rest zero)
#define WS_WPOW 768                    // NPOW_TOT * 100 floats: W^(2^j)
#define WS_AGG  (WS_WPOW + NPOW_TOT * 100)          // [ch][seg][10] segment aggregates
// WS_CARRY = WS_AGG + NCH*segs*10                  // [ch][seg][10] segment entry states

struct Biq { float b0, b1, b2, a1, a2; };

__device__ __host__ inline float step_cascade(const Biq* f, float* s, float x) {
  float v = x;
#pragma unroll
  for (int i = 0; i < 5; i++) {
    float y      = f[i].b0 * v + s[2 * i];
    s[2 * i]     = f[i].b1 * v - f[i].a1 * y + s[2 * i + 1];
    s[2 * i + 1] = f[i].b2 * v - f[i].a2 * y;
    v = y;
  }
  return v;
}

// ---------------------------------------------------------------------------
// Setup (1 thread): biquad coeffs from eq_params, exact block matrices
// T, Vpad, Upad by impulse simulation, and the W^(2^j) power chain.
// ---------------------------------------------------------------------------
__global__ void eq_setup_kernel(const float* __restrict__ eqp, float* __restrict__ ws) {
  if (threadIdx.x != 0 || blockIdx.x != 0) return;
  const float sr = 44100.0f;
  Biq f[5];
  for (int i = 0; i < 5; i++) {
    float g = eqp[i * 3 + 0], fc = eqp[i * 3 + 1], q = eqp[i * 3 + 2];
    float A  = powf(10.0f, g / 40.0f);
    float w0 = 2.0f * 3.14159265358979323846f * (fc / sr);
    float al = sinf(w0) / (2.0f * q);
    float c  = cosf(w0);
    float b0, b1, b2, a0, a1, a2;
    if (i == 0 || i == 4) {                 // shelves (i==0 low, i==4 high)
      float sgn = (i == 4) ? 1.0f : -1.0f;
      float sA  = sqrtf(A);
      b0 = A * ((A + 1.0f) + sgn * (A - 1.0f) * c + 2.0f * sA * al);
      b1 = -2.0f * sgn * A * ((A - 1.0f) + sgn * (A + 1.0f) * c);
      b2 = A * ((A + 1.0f) + sgn * (A - 1.0f) * c - 2.0f * sA * al);
      a0 = (A + 1.0f) - sgn * (A - 1.0f) * c + 2.0f * sA * al;
      a1 = 2.0f * sgn * ((A - 1.0f) - sgn * (A + 1.0f) * c);
      a2 = (A + 1.0f) - sgn * (A - 1.0f) * c - 2.0f * sA * al;
    } else {                                // peaking
      b0 = 1.0f + al * A; b1 = -2.0f * c; b2 = 1.0f - al * A;
      a0 = 1.0f + al / A; a1 = -2.0f * c; a2 = 1.0f - al / A;
    }
    f[i].b0 = b0 / a0; f[i].b1 = b1 / a0; f[i].b2 = b2 / a0;
    f[i].a1 = a1 / a0; f[i].a2 = a2 / a0;
  }
  // Columns of T (zero-state response) and V (final state) from unit impulses
  for (int i = 0; i < 16; i++) {
    float s[NSTATE];
    for (int k = 0; k < NSTATE; k++) s[k] = 0.0f;
    for (int t = 0; t < 16; t++)
      ws[WS_T + t * 16 + i] = step_cascade(f, s, (t == i) ? 1.0f : 0.0f);
    for (int k = 0; k < NSTATE; k++) ws[WS_V + k * 16 + i] = s[k];
    for (int k = NSTATE; k < 16; k++) ws[WS_V + k * 16 + i] = 0.0f;
  }
  // Columns of U (state->output) and W (state->state) from unit states
  float W[NSTATE][NSTATE];
  for (int j = 0; j < 16; j++) {
    float s[NSTATE];
    for (int k = 0; k < NSTATE; k++) s[k] = (k == j) ? 1.0f : 0.0f;
    for (int t = 0; t < 16; t++) {
      float y = step_cascade(f, s, 0.0f);
      ws[WS_U + t * 16 + j] = (j < NSTATE) ? y : 0.0f;
    }
    if (j < NSTATE)
      for (int k = 0; k < NSTATE; k++) W[k][j] = s[k];
  }
  // W^(2^j), j = 0..NPOW_TOT-1 (up to W^2048 for segment linking)
  float* wp = ws + WS_WPOW;
  for (int c = 0; c < NSTATE; c++)
    for (int d = 0; d < NSTATE; d++) wp[c * 10 + d] = W[c][d];
  for (int p = 1; p < NPOW_TOT; p++) {
    const float* a = wp + (p - 1) * 100;
    float* o = wp + p * 100;
    for (int c = 0; c < NSTATE; c++)
      for (int d = 0; d < NSTATE; d++) {
        float acc = 0.0f;
        for (int e = 0; e < NSTATE; e++) acc += a[c * 10 + e] * a[e * 10 + d];
        o[c * 10 + d] = acc;
      }
  }
}

// ---------------------------------------------------------------------------
// Shared device helpers
// ---------------------------------------------------------------------------

// Scatter a V*X WMMA result (D layout: VGPR j = row M=j for lanes<16, M=j+8
// for lanes>=16; N = lane&15) into the LDS state array rows 1..512.
__device__ inline void scatter_vt(float st[][ST_STRIDE], const v8f& vt,
                                  int wave, int n, int h) {
  int row = 1 + wave * TILE + n;
  if (h == 0) {
#pragma unroll
    for (int j = 0; j < 8; j++) st[row][j] = vt[j];
  } else {
    st[row][8] = vt[0];
    st[row][9] = vt[1];
  }
}

// Kogge-Stone inclusive scan of t_i = v_i + W * t_{i-1} over 512 block-state
// vectors in LDS; carry folded in via v_0 += W * st[0]. Leaves the
// chunk carry-out in st[512] (caller copies it to st[0]). Must be called by all THREADS threads.
__device__ inline void scan_states(float st[][ST_STRIDE], const float* sW, int tid) {
  float t[NSTATE];
  const bool active = (tid < BLOCKS_PER_CHUNK);
  if (active) {
#pragma unroll
    for (int c = 0; c < NSTATE; c++) t[c] = st[tid + 1][c];
    if (tid == 0) {
#pragma unroll
      for (int c = 0; c < NSTATE; c++) {
        float acc = t[c];
        for (int d = 0; d < NSTATE; d++) acc += sW[c * 10 + d] * st[0][d];
        t[c] = acc;
      }
#pragma unroll
      for (int c = 0; c < NSTATE; c++) st[1][c] = t[c];
    }
  }
  __syncthreads();
#pragma unroll 1
  for (int j = 0; j < NPOW; j++) {
    const int k = 1 << j;
    float nb[NSTATE];
    const bool doit = active && (tid >= k);
    if (doit) {
#pragma unroll
      for (int c = 0; c < NSTATE; c++) nb[c] = st[tid + 1 - k][c];
    }
    __syncthreads();
    if (doit) {
      const float* Wk = sW + j * 100;
#pragma unroll
      for (int c = 0; c < NSTATE; c++) {
        float acc = t[c];
        for (int d = 0; d < NSTATE; d++) acc += Wk[c * 10 + d] * nb[d];
        t[c] = acc;
        st[tid + 1][c] = acc;
      }
    }
    __syncthreads();
  }
}

__device__ inline void load_lds_common(float st[][ST_STRIDE], float* sW,
                                       const float* ws, int tid) {
  for (int i = tid; i < NPOW * 100; i += THREADS) sW[i] = ws[WS_WPOW + i];
  // Zero the whole state array (K=10..15 of padded B operands must read zeros
  // so padded A columns never see 0*garbage).
  for (int i = tid; i < (BLOCKS_PER_CHUNK + 1) * ST_STRIDE; i += THREADS)
    ((float*)st)[i] = 0.0f;
}

// ---------------------------------------------------------------------------
// Pass A: per (segment, channel) WG, compute the segment aggregate
// a_seg = contribution of the segment's inputs to its exit state (zero entry
// state). Only the V*X WMMAs + scan are needed; nothing is written but a_seg.
// ---------------------------------------------------------------------------
__global__ __launch_bounds__(THREADS) void eq_passA_kernel(
    const float* __restrict__ x, float* __restrict__ ws, int nsamp, int segs) {
  __shared__ float st[BLOCKS_PER_CHUNK + 1][ST_STRIDE];
  __shared__ float sW[NPOW * 100];

  const int tid  = threadIdx.x;
  const int wave = tid >> 5;
  const int lane = tid & 31;
  const int n = lane & 15, h = lane >> 4;
  const int seg = blockIdx.x, ch = blockIdx.y;

  const float* xc = x + (size_t)ch * (size_t)nsamp + (size_t)seg * SEG_SAMPS;

  load_lds_common(st, sW, ws, tid);

  v2f Va[4];
#pragma unroll
  for (int s = 0; s < 4; s++)
    Va[s] = *(const v2f*)(ws + WS_V + n * 16 + (4 * s + 2 * h));
  __syncthreads();

  const int tileoff = wave * (TILE * BLK) + n * BLK;
  v2f Xb[4];
#pragma unroll
  for (int s = 0; s < 4; s++)
    Xb[s] = *(const v2f*)(xc + tileoff + (4 * s + 2 * h));

#pragma unroll 1
  for (int q = 0; q < CHUNKS_PER_SEG; q++) {
    v8f vt = {};
#pragma unroll
    for (int s = 0; s < 4; s++)
      vt = __builtin_amdgcn_wmma_f32_16x16x4_f32(false, Va[s], false, Xb[s],
                                                 (short)0, vt, false, false);
    // Pipeline next chunk's X tile under the scan.
    const int qn = (q + 1 < CHUNKS_PER_SEG) ? (q + 1) : q;
    v2f Xn[4];
#pragma unroll
    for (int s = 0; s < 4; s++)
      Xn[s] = *(const v2f*)(xc + qn * CHUNK_SAMPS + tileoff + (4 * s + 2 * h));

    scatter_vt(st, vt, wave, n, h);
    __syncthreads();
    scan_states(st, sW, tid);
    if (tid < NSTATE) st[0][tid] = st[BLOCKS_PER_CHUNK][tid];  // carry out
    __syncthreads();
#pragma unroll
    for (int s = 0; s < 4; s++) Xb[s] = Xn[s];
  }
  if (tid < NSTATE)
    ws[WS_AGG + ((size_t)ch * segs + seg) * NSTATE + tid] = st[0][tid];
}

// ---------------------------------------------------------------------------
// Pass B: per channel, link segments: c_0 = 0; c_{s+1} = W^2048 c_s + a_s.
// Writes the exact entry state of every segment.
// ---------------------------------------------------------------------------
__global__ void eq_passB_kernel(float* __restrict__ ws, int segs) {
  const int ch = threadIdx.x;
  if (ch >= NCH || blockIdx.x != 0) return;
  const float* W2048 = ws + WS_WPOW + (NPOW_TOT - 1) * 100;
  const float* agg = ws + WS_AGG;
  float* carry = ws + WS_AGG + (size_t)NCH * segs * NSTATE;
  float c[NSTATE];
  for (int k = 0; k < NSTATE; k++) c[k] = 0.0f;
  for (int s = 0; s < segs; s++) {
    const size_t off = ((size_t)ch * segs + s) * NSTATE;
    float nc[NSTATE];
    for (int i = 0; i < NSTATE; i++) {
      float acc = agg[off + i];
      for (int d = 0; d < NSTATE; d++) acc += W2048[i * 10 + d] * c[d];
      carry[off + i] = c[i];
      nc[i] = acc;
    }
    for (int i = 0; i < NSTATE; i++) c[i] = nc[i];
  }
}

// ---------------------------------------------------------------------------
// Pass C: per (segment, channel) WG, full pipeline seeded with the exact
// segment entry state: Y0 = T*X, scan of V*X states, Y = Y0 + Upad*S (fused
// via the WMMA C accumulator), coalesced b128 stores.
// ---------------------------------------------------------------------------
__global__ __launch_bounds__(THREADS) void eq_passC_kernel(
    const float* __restrict__ x, const float* __restrict__ ws,
    float* __restrict__ out, int nsamp, int segs) {
  __shared__ float st[BLOCKS_PER_CHUNK + 1][ST_STRIDE];
  __shared__ float sW[NPOW * 100];

  const int tid  = threadIdx.x;
  const int wave = tid >> 5;
  const int lane = tid & 31;
  const int n = lane & 15, h = lane >> 4;
  const int seg = blockIdx.x, ch = blockIdx.y;

  const float* xc = x + (size_t)ch * (size_t)nsamp + (size_t)seg * SEG_SAMPS;
  float*       oc = out + (size_t)ch * (size_t)nsamp + (size_t)seg * SEG_SAMPS;

  load_lds_common(st, sW, ws, tid);

  v2f Ta[4], Va[4], Ua[4];
#pragma unroll
  for (int s = 0; s < 4; s++) {
    int k = 4 * s + 2 * h;
    Ta[s] = *(const v2f*)(ws + WS_T + n * 16 + k);
    Va[s] = *(const v2f*)(ws + WS_V + n * 16 + k);
    Ua[s] = *(const v2f*)(ws + WS_U + n * 16 + k);
  }
  // Seed st[0] with this segment's exact entry state (from pass B).
  const float* carry = ws + WS_AGG + (size_t)NCH * segs * NSTATE;
  if (tid < NSTATE)
    st[0][tid] = carry[((size_t)ch * segs + seg) * NSTATE + tid];
  __syncthreads();

  const int tileoff = wave * (TILE * BLK) + n * BLK;
  v2f Xb[4];
#pragma unroll
  for (int s = 0; s < 4; s++)
    Xb[s] = *(const v2f*)(xc + tileoff + (4 * s + 2 * h));

#pragma unroll 1
  for (int q = 0; q < CHUNKS_PER_SEG; q++) {
    const int base = q * CHUNK_SAMPS + wave * (TILE * BLK);

    v8f y  = {};
    v8f vt = {};
#pragma unroll
    for (int s = 0; s < 4; s++) {
      y  = __builtin_amdgcn_wmma_f32_16x16x4_f32(false, Ta[s], false, Xb[s],
                                                 (short)0, y, false, false);
      vt = __builtin_amdgcn_wmma_f32_16x16x4_f32(false, Va[s], false, Xb[s],
                                                 (short)0, vt, false, false);
    }
    const int qn = (q + 1 < CHUNKS_PER_SEG) ? (q + 1) : q;
    v2f Xn[4];
#pragma unroll
    for (int s = 0; s < 4; s++)
      Xn[s] = *(const v2f*)(xc + qn * CHUNK_SAMPS + tileoff + (4 * s + 2 * h));

    scatter_vt(st, vt, wave, n, h);
    __syncthreads();
    scan_states(st, sW, tid);

    // S column n = state BEFORE block n = st[wave*16 + n] (st[0] = carry-in).
    v2f Sb[4];
#pragma unroll
    for (int s = 0; s < 4; s++)
      Sb[s] = *(const v2f*)(&st[wave * TILE + n][4 * s + 2 * h]);
#pragma unroll
    for (int s = 0; s < 4; s++)
      y = __builtin_amdgcn_wmma_f32_16x16x4_f32(false, Ua[s], false, Sb[s],
                                                (short)0, y, false, false);

    float* p = oc + base + n * BLK + 8 * h;
    *(float4*)(p)     = make_float4(y[0], y[1], y[2], y[3]);
    *(float4*)(p + 4) = make_float4(y[4], y[5], y[6], y[7]);

    __syncthreads();
    if (tid < NSTATE) st[0][tid] = st[BLOCKS_PER_CHUNK][tid];  // carry out
    __syncthreads();
#pragma unroll
    for (int s = 0; s < 4; s++) Xb[s] = Xn[s];
  }
}

extern "C" void kernel_launch(void* const* d_in, const int* in_sizes, int n_in,
                              void* d_out, int out_size, void* d_ws, size_t ws_size,
                              hipStream_t stream) {
  const float* x   = (const float*)d_in[0];
  const float* eqp = (const float*)d_in[1];
  float* out = (float*)d_out;
  float* ws  = (float*)d_ws;
  const int nsamp = in_sizes[0] / NCH;
  const int segs  = nsamp / SEG_SAMPS;   // 64 for 2M samples

  eq_setup_kernel<<<1, 32, 0, stream>>>(eqp, ws);
  eq_passA_kernel<<<dim3(segs, NCH), THREADS, 0, stream>>>(x, ws, nsamp, segs);
  eq_passB_kernel<<<1, 32, 0, stream>>>(ws, segs);
  eq_passC_kernel<<<dim3(segs, NCH), THREADS, 0, stream>>>(x, ws, out, nsamp, segs);
}